// DWTLevel_24197845746131
// MI455X (gfx1250) — compile-verified
//
#include <hip/hip_runtime.h>
#include <cstdint>

typedef float v4f __attribute__((ext_vector_type(4)));

#define WAVESZ      32
#define WPB         8                  // waves per block
#define BLOCK       (WAVESZ * WPB)     // 256 threads
#define TASK_FLOATS 1024               // two contiguous input rows of 512 fp32 = 4 KiB

// Issue 8 x 512B async b128 loads: one contiguous 4 KiB task tile -> LDS.
// GVS addressing: mem = saddr + vaddr + offset ; LDS dst = vdst + offset.
// The same immediate offset is applied to BOTH the global and LDS address,
// so one (ldsaddr, voff) pair covers the whole tile.
__device__ __forceinline__ void async_load_task(unsigned ldsaddr, unsigned voff,
                                                unsigned long long sbase) {
  asm volatile(
      "s_wait_dscnt 0x0\n\t"  // all LDS reads of the buffer being overwritten are done
      "global_load_async_to_lds_b128 %0, %1, %2 offset:0    th:TH_LOAD_NT\n\t"
      "global_load_async_to_lds_b128 %0, %1, %2 offset:512  th:TH_LOAD_NT\n\t"
      "global_load_async_to_lds_b128 %0, %1, %2 offset:1024 th:TH_LOAD_NT\n\t"
      "global_load_async_to_lds_b128 %0, %1, %2 offset:1536 th:TH_LOAD_NT\n\t"
      "global_load_async_to_lds_b128 %0, %1, %2 offset:2048 th:TH_LOAD_NT\n\t"
      "global_load_async_to_lds_b128 %0, %1, %2 offset:2560 th:TH_LOAD_NT\n\t"
      "global_load_async_to_lds_b128 %0, %1, %2 offset:3072 th:TH_LOAD_NT\n\t"
      "global_load_async_to_lds_b128 %0, %1, %2 offset:3584 th:TH_LOAD_NT"
      :: "v"(ldsaddr), "v"(voff), "s"(sbase) : "memory");
}

__device__ __forceinline__ void wait_async_le8() {
  asm volatile("s_wait_asynccnt 0x8" ::: "memory");   // previous tile's 8 loads done
}
__device__ __forceinline__ void wait_async_0() {
  asm volatile("s_wait_asynccnt 0x0" ::: "memory");
}

__global__ __launch_bounds__(BLOCK) void haar_dwt2_kernel(
    const float* __restrict__ x, float* __restrict__ out, int num_tasks) {
  __shared__ float lds[WPB * 2 * TASK_FLOATS];  // 64 KiB: 8 waves x double buffer x 4 KiB

  const int lane   = threadIdx.x & (WAVESZ - 1);
  const int wib    = threadIdx.x >> 5;                // wave in block
  const int wave   = blockIdx.x * WPB + wib;
  const int nwaves = gridDim.x * WPB;

  // Generic pointers for LDS reads; raw 32-bit LDS byte addresses for the async asm.
  float* pA = &lds[wib * 2 * TASK_FLOATS];
  float* pB = pA + TASK_FLOATS;
  const unsigned voff = (unsigned)(lane * 16);        // per-lane 16B within each 512B chunk
  unsigned bufA = (unsigned)(uintptr_t)pA + voff;
  unsigned bufB = (unsigned)(uintptr_t)pB + voff;

  if (wave >= num_tasks) return;

  const size_t Ncomp = (size_t)num_tasks * 256;       // floats per sub-band

  long t = wave;
  async_load_task(bufA, voff, (unsigned long long)(x + (size_t)t * TASK_FLOATS));

  unsigned ldsCur = bufA, ldsNxt = bufB;
  const float* pCur = pA;
  const float* pNxt = pB;

  for (; t < num_tasks; t += nwaves) {
    const long tn = t + nwaves;
    const bool have_next = (tn < (long)num_tasks);
    if (have_next) {
      async_load_task(ldsNxt, voff,
                      (unsigned long long)(x + (size_t)tn * TASK_FLOATS));
      wait_async_le8();
    } else {
      wait_async_0();
    }

    // Read this lane's 2 x 16 input floats from LDS (ds_load_b128 x 8).
    float r0f[16] __attribute__((aligned(16)));
    float r1f[16] __attribute__((aligned(16)));
    const float* base = pCur + lane * 16;
#pragma unroll
    for (int q = 0; q < 4; ++q) {
      *(v4f*)&r0f[4 * q] = *(const v4f*)(base + 4 * q);          // row 2i
      *(v4f*)&r1f[4 * q] = *(const v4f*)(base + 512 + 4 * q);    // row 2i+1
    }

    float ll[8] __attribute__((aligned(16)));
    float lh[8] __attribute__((aligned(16)));
    float hl[8] __attribute__((aligned(16)));
    float hh[8] __attribute__((aligned(16)));
#pragma unroll
    for (int j = 0; j < 8; ++j) {
      const float a = r0f[2 * j], b = r0f[2 * j + 1];
      const float c = r1f[2 * j], d = r1f[2 * j + 1];
      const float apb = a + b, amb = a - b;
      const float cpd = c + d, cmd = c - d;
      ll[j] = (apb + cpd) * 0.5f;
      lh[j] = (apb - cpd) * 0.5f;
      hl[j] = (amb + cmd) * 0.5f;
      hh[j] = (amb - cmd) * 0.5f;
    }

    // Streaming NT stores: 2 x b128 per sub-band per lane.
    const size_t obase = (size_t)t * 256 + (size_t)lane * 8;
    float* p0 = out + obase;                 // LL
    float* p1 = out + Ncomp + obase;         // LH
    float* p2 = out + 2 * Ncomp + obase;     // HL
    float* p3 = out + 3 * Ncomp + obase;     // HH
    __builtin_nontemporal_store(*(v4f*)&ll[0], (v4f*)p0);
    __builtin_nontemporal_store(*(v4f*)&ll[4], (v4f*)(p0 + 4));
    __builtin_nontemporal_store(*(v4f*)&lh[0], (v4f*)p1);
    __builtin_nontemporal_store(*(v4f*)&lh[4], (v4f*)(p1 + 4));
    __builtin_nontemporal_store(*(v4f*)&hl[0], (v4f*)p2);
    __builtin_nontemporal_store(*(v4f*)&hl[4], (v4f*)(p2 + 4));
    __builtin_nontemporal_store(*(v4f*)&hh[0], (v4f*)p3);
    __builtin_nontemporal_store(*(v4f*)&hh[4], (v4f*)(p3 + 4));

    // swap double buffers
    unsigned tl = ldsCur; ldsCur = ldsNxt; ldsNxt = tl;
    const float* tp = pCur; pCur = pNxt; pNxt = tp;
  }
}

extern "C" void kernel_launch(void* const* d_in, const int* in_sizes, int n_in,
                              void* d_out, int out_size, void* d_ws, size_t ws_size,
                              hipStream_t stream) {
  (void)n_in; (void)out_size; (void)d_ws; (void)ws_size;
  const float* x = (const float*)d_in[0];
  float* out = (float*)d_out;

  // x: (8, 64, 512, 512) fp32. One task = one output row = two contiguous
  // input rows = 1024 floats. num_tasks = total / 1024 = 131072.
  const int total = in_sizes[0];
  const int num_tasks = total / TASK_FLOATS;

  int grid = (num_tasks + WPB - 1) / WPB;
  if (grid > 2048) grid = 2048;   // 16384 waves -> 8 tasks/wave for async pipelining
  if (grid < 1) grid = 1;

  haar_dwt2_kernel<<<grid, BLOCK, 0, stream>>>(x, out, num_tasks);
}